// group_change_14448269984373
// MI455X (gfx1250) — compile-verified
//
#include <hip/hip_runtime.h>
#include <math.h>

// ---- problem constants (from reference) ----
#define P    128      // rows per window (p)
#define DD   96       // feature dim D
#define GCHN 101      // group channels (SIM+1)
#define SIMK 100      // top-k count
#define LCHN 80       // latent channels
#define NWIN 64       // N*W windows
#define SV   100      // padded LDS stride for 96-wide mats (16B aligned, conflict-free)
#define SS   132      // padded LDS stride for 128-wide mats (16B aligned, conflict-free)

typedef float v2f __attribute__((ext_vector_type(2)));
typedef float v8f __attribute__((ext_vector_type(8)));
typedef unsigned int u32x4 __attribute__((ext_vector_type(4)));
typedef int i32x4 __attribute__((ext_vector_type(4)));
typedef int i32x8 __attribute__((ext_vector_type(8)));

__device__ __forceinline__ v8f wmma4(v2f a, v2f b, v8f c) {
    // D(16x16 f32) = A(16x4 f32) * B(4x16 f32) + C
    return __builtin_amdgcn_wmma_f32_16x16x4_f32(
        /*neg_a=*/false, a, /*neg_b=*/false, b,
        /*c_mod=*/(short)0, c, /*reuse_a=*/false, /*reuse_b=*/false);
}

// C_tile(16x16) at (row0,n0) of  A[P x K] * B^T, with B stored row-major [N][K]
template<int K>
__device__ __forceinline__ v8f gemm_tile_abt(const float* A, int lda,
                                             const float* B, int ldb,
                                             int row0, int n0, int lane) {
    v8f c = {0.f,0.f,0.f,0.f,0.f,0.f,0.f,0.f};
    const int half = lane >> 4;
    const int lr   = lane & 15;
    const float* ap = A + (row0 + lr) * lda + 2 * half;
    const float* bp = B + (n0  + lr) * ldb + 2 * half;
    #pragma unroll
    for (int k0 = 0; k0 < K; k0 += 4) {
        v2f a; a.x = ap[k0]; a.y = ap[k0 + 1];
        v2f b; b.x = bp[k0]; b.y = bp[k0 + 1];
        c = wmma4(a, b, c);
    }
    return c;
}

// C_tile(16x16) at (row0,n0) of  A[P x K] * B, with B stored row-major [K][N]
template<int K>
__device__ __forceinline__ v8f gemm_tile_ab(const float* A, int lda,
                                            const float* B, int ldb,
                                            int row0, int n0, int lane) {
    v8f c = {0.f,0.f,0.f,0.f,0.f,0.f,0.f,0.f};
    const int half = lane >> 4;
    const int lr   = lane & 15;
    const float* ap = A + (row0 + lr) * lda + 2 * half;
    const float* bp = B + (n0 + lr);
    #pragma unroll
    for (int k0 = 0; k0 < K; k0 += 4) {
        v2f a; a.x = ap[k0]; a.y = ap[k0 + 1];
        v2f b; b.x = bp[(k0 + 2 * half) * ldb];
               b.y = bp[(k0 + 2 * half + 1) * ldb];
        c = wmma4(a, b, c);
    }
    return c;
}

__device__ __forceinline__ void store_tile(float* C, int ldc, int row0, int n0,
                                           v8f c, int lane) {
    const int half = lane >> 4;
    const int lr   = lane & 15;
    #pragma unroll
    for (int j = 0; j < 8; j++)
        C[(row0 + 8 * half + j) * ldc + n0 + lr] = c[j];
}

__device__ __forceinline__ float wave_max32(float v) {
    #pragma unroll
    for (int o = 16; o >= 1; o >>= 1) v = fmaxf(v, __shfl_xor(v, o, 32));
    return v;
}
__device__ __forceinline__ float wave_sum32(float v) {
    #pragma unroll
    for (int o = 16; o >= 1; o >>= 1) v += __shfl_xor(v, o, 32);
    return v;
}

// stage a ROWS x 96 row-major f32 matrix from global into stride-SV LDS via b128
template<int ROWS>
__device__ __forceinline__ void stage96(const float* __restrict__ g,
                                        float* s, int tid) {
    const float4* g4 = (const float4*)g;
    #pragma unroll 4
    for (int i = tid; i < ROWS * 24; i += 256) {
        int r = i / 24, c4 = i % 24;
        ((float4*)(s + r * SV))[c4] = g4[i];
    }
}

// ---------------------------------------------------------------------------
// Kernel 1: the constant per-group-channel scale vector.
// softmax over a singleton axis == 1, so scale[g] = sigmoid(mb . w_up[g] + b_up[g])
// ---------------------------------------------------------------------------
__global__ void scale_kernel(const float* __restrict__ mb,
                             const float* __restrict__ w_up,
                             const float* __restrict__ b_up,
                             float* __restrict__ scale) {
    int g = threadIdx.x;
    if (g < GCHN) {
        float s = b_up[g];
        #pragma unroll 8
        for (int j = 0; j < LCHN; j++) s += mb[j] * w_up[g * LCHN + j];
        scale[g] = 1.0f / (1.0f + __expf(-s));
    }
}

// ---------------------------------------------------------------------------
// Kernel 2: one block per window.  8 waves x 32 lanes.
// ---------------------------------------------------------------------------
__global__ void __launch_bounds__(256)
window_kernel(const float* __restrict__ attn,
              const float* __restrict__ v,
              const float* __restrict__ scale_g,
              const float* __restrict__ w_q, const float* __restrict__ b_q,
              const float* __restrict__ w_k, const float* __restrict__ b_k,
              const float* __restrict__ w_o, const float* __restrict__ b_o,
              float* __restrict__ out) {
    __shared__ __align__(16) float s_vq [P * SV];   // vq (128x96), whole kernel
    __shared__ __align__(16) float s_big[P * SS];   // attn -> S -> Wdense
    __shared__ __align__(16) float s_Q  [P * SV];   // Q -> O
    __shared__ __align__(16) float s_K  [P * SV];   // K0
    __shared__ __align__(16) float s_w  [DD * SV];  // staged weight (w_q->w_k->w_o)
    __shared__ unsigned char s_slot[P * P];         // sorted-slot or 0xFF
    __shared__ float s_scale[GCHN];
    __shared__ float s_qb[P];

    const int tid  = threadIdx.x;
    const int wave = tid >> 5;
    const int lane = tid & 31;
    const int win  = blockIdx.x;

    const float* attnw = attn + (size_t)win * P * P;
    const float* vw    = v    + (size_t)win * P * DD;

    // ---- stage attn via Tensor Data Mover: 128x128 f32 tile, LDS-padded to
    //      stride 132 dwords (pad 4 dwords after every 128) ----
#if __has_builtin(__builtin_amdgcn_tensor_load_to_lds) && __has_builtin(__builtin_amdgcn_s_wait_tensorcnt)
    if (wave == 0) {
        unsigned lds_addr = (unsigned)(size_t)(void*)s_big;  // low 32b = LDS offset
        unsigned long long ga = (unsigned long long)(size_t)attnw;
        u32x4 g0;
        g0.x = 1u;                                        // count=1, no gather
        g0.y = lds_addr;                                  // lds_addr [63:32]
        g0.z = (unsigned)(ga & 0xffffffffu);              // global_addr [95:64]
        g0.w = ((unsigned)(ga >> 32) & 0x01ffffffu)       // global_addr [120:96]
             | (2u << 30);                                // type=2 (image)
        i32x8 g1;
        g1[0] = (2 << 16)      // data_size = 4B
              | (1 << 20)      // pad_enable
              | (6 << 22)      // pad_interval: every 128 dwords
              | (3 << 25);     // pad_amount: 4 dwords
        g1[1] = (int)(128u << 16);   // tensor_dim0 = 128   (bits 79:48)
        g1[2] = (int)(128u << 16);   // tensor_dim1 = 128   (bits 111:80)
        g1[3] = (int)(128u << 16);   // tile_dim0   = 128   (bits 127:112)
        g1[4] = 128;                 // tile_dim1   = 128   (bits 143:128)
        g1[5] = 128;                 // tensor_dim0_stride  (bits 207:160)
        g1[6] = 0;
        g1[7] = 0;
        i32x4 z4 = {0, 0, 0, 0};         // 2D tensor: groups 2/3 unused
        i32x8 z8 = {0, 0, 0, 0, 0, 0, 0, 0};
        __builtin_amdgcn_tensor_load_to_lds(g0, g1, z4, z4, z8, 0);
    }
#else
    for (int i = tid; i < P * P; i += 256)
        s_big[(i >> 7) * SS + (i & 127)] = attnw[i];
#endif

    // ---- stage vq, w_q, scale with b128 loads (overlaps the TDM) ----
    stage96<P>(vw, s_vq, tid);
    stage96<DD>(w_q, s_w, tid);
    if (tid < GCHN) s_scale[tid] = scale_g[tid];

#if __has_builtin(__builtin_amdgcn_tensor_load_to_lds) && __has_builtin(__builtin_amdgcn_s_wait_tensorcnt)
    if (wave == 0) __builtin_amdgcn_s_wait_tensorcnt(0);
#endif
    __syncthreads();

    // ---- Phase A: per-row top-100 selection + sorted-slot assignment ----
    // keep(r) iff fewer than 100 elements rank above it (value desc, index asc).
    // slot(r) = #kept indices < r  (idx is sorted ascending in the reference).
    for (int p = wave; p < P; p += 8) {
        const float* row = &s_big[p * SS];
        int   rr[4]; float av[4]; int cnt[4];
        #pragma unroll
        for (int i = 0; i < 4; i++) {
            rr[i] = lane + 32 * i; av[i] = row[rr[i]]; cnt[i] = 0;
        }
        for (int j = 0; j < P; j++) {
            float vj = row[j];
            #pragma unroll
            for (int i = 0; i < 4; i++)
                cnt[i] += (vj > av[i]) || (vj == av[i] && j < rr[i]);
        }
        unsigned m[4];
        #pragma unroll
        for (int i = 0; i < 4; i++)
            m[i] = (unsigned)__ballot(cnt[i] < SIMK);
        const unsigned lt = (1u << lane) - 1u;
        int base = 0;
        #pragma unroll
        for (int i = 0; i < 4; i++) {
            int slot = base + __popc(m[i] & lt);
            s_slot[p * P + rr[i]] =
                (cnt[i] < SIMK) ? (unsigned char)slot : (unsigned char)0xFF;
            base += __popc(m[i]);
        }
    }

    const int rows0 = wave * 16;

    // ---- Q = scale0*(vq @ w_q^T) + b_q  (w_q staged in LDS) ----
    {
        const float sc0 = s_scale[0];
        const int half = lane >> 4, lr = lane & 15;
        #pragma unroll
        for (int t = 0; t < 6; t++) {
            int n0 = t * 16;
            v8f acc = gemm_tile_abt<DD>(s_vq, SV, s_w, SV, rows0, n0, lane);
            float bq = b_q[n0 + lr];
            #pragma unroll
            for (int j = 0; j < 8; j++)
                s_Q[(rows0 + 8 * half + j) * SV + n0 + lr] = sc0 * acc[j] + bq;
        }
    }
    __syncthreads();

    // ---- swap staged weight -> w_k ----
    stage96<DD>(w_k, s_w, tid);
    __syncthreads();

    // ---- K0 = vq @ w_k^T  (bias folded into qb) ----
    #pragma unroll
    for (int t = 0; t < 6; t++) {
        int n0 = t * 16;
        v8f acc = gemm_tile_abt<DD>(s_vq, SV, s_w, SV, rows0, n0, lane);
        store_tile(s_K, SV, rows0, n0, acc, lane);
    }
    __syncthreads();

    // ---- swap staged weight -> w_o (consumed after 3 more barriers) ----
    stage96<DD>(w_o, s_w, tid);

    // ---- qb[p] = Q[p] . b_k ----
    if (tid < P) {
        float s = 0.f;
        #pragma unroll 8
        for (int c2 = 0; c2 < DD; c2++) s += s_Q[tid * SV + c2] * b_k[c2];
        s_qb[tid] = s;
    }

    // ---- S = Q @ K0^T  (dense 128x128, overwrites attn region) ----
    #pragma unroll
    for (int t = 0; t < 8; t++) {
        int r0 = t * 16;
        v8f acc = gemm_tile_abt<DD>(s_Q, SV, s_K, SV, rows0, r0, lane);
        store_tile(s_big, SS, rows0, r0, acc, lane);
    }
    __syncthreads();

    // ---- masked softmax over selected entries; build Wdense in place ----
    // score(p,s) = (scale[1+slot]*S[p,r] + qb[p]) / sqrt(96)
    // Wdense[p,r] = softmax * scale[1+slot]   (out uses yk = scale*vq, no b_k)
    const float rsq = 0.1020620726159658f; // 1/sqrt(96)
    for (int p = wave; p < P; p += 8) {
        float* row = &s_big[p * SS];
        const float qbp = s_qb[p];
        float val[4], scv[4];
        #pragma unroll
        for (int i = 0; i < 4; i++) {
            int r = lane + 32 * i;
            unsigned char sl = s_slot[p * P + r];
            if (sl != 0xFF) {
                float sc = s_scale[1 + sl];
                scv[i] = sc;
                val[i] = (sc * row[r] + qbp) * rsq;
            } else {
                scv[i] = 0.f;
                val[i] = -3.0e38f;
            }
        }
        float m = fmaxf(fmaxf(val[0], val[1]), fmaxf(val[2], val[3]));
        m = wave_max32(m);
        float e[4], s = 0.f;
        #pragma unroll
        for (int i = 0; i < 4; i++) {
            e[i] = (val[i] > -1.0e38f) ? __expf(val[i] - m) : 0.f;
            s += e[i];
        }
        s = wave_sum32(s);
        float inv = 1.0f / s;
        #pragma unroll
        for (int i = 0; i < 4; i++)
            row[lane + 32 * i] = e[i] * inv * scv[i];
    }
    __syncthreads();

    // ---- O = Wdense @ vq  (128x128 * 128x96), overwrites Q region ----
    #pragma unroll
    for (int t = 0; t < 6; t++) {
        int n0 = t * 16;
        v8f acc = gemm_tile_ab<P>(s_big, SS, s_vq, SV, rows0, n0, lane);
        store_tile(s_Q, SV, rows0, n0, acc, lane);
    }
    __syncthreads();

    // ---- R = O @ w_o^T + b_o -> global ----
    {
        const int half = lane >> 4, lr = lane & 15;
        #pragma unroll
        for (int t = 0; t < 6; t++) {
            int n0 = t * 16;
            v8f acc = gemm_tile_abt<DD>(s_Q, SV, s_w, SV, rows0, n0, lane);
            float bo = b_o[n0 + lr];
            #pragma unroll
            for (int j = 0; j < 8; j++) {
                int grow = win * P + rows0 + 8 * half + j;
                out[(size_t)grow * DD + n0 + lr] = acc[j] + bo;
            }
        }
    }
}

extern "C" void kernel_launch(void* const* d_in, const int* in_sizes, int n_in,
                              void* d_out, int out_size, void* d_ws, size_t ws_size,
                              hipStream_t stream) {
    // setup_inputs order:
    // 0 attn, 1 v, 2 mb, 3 w_sub, 4 b_sub, 5 w_up, 6 b_up,
    // 7 w_q, 8 b_q, 9 w_k, 10 b_k, 11 w_o, 12 b_o
    const float* attn = (const float*)d_in[0];
    const float* v    = (const float*)d_in[1];
    const float* mb   = (const float*)d_in[2];
    const float* w_up = (const float*)d_in[5];
    const float* b_up = (const float*)d_in[6];
    const float* w_q  = (const float*)d_in[7];
    const float* b_q  = (const float*)d_in[8];
    const float* w_k  = (const float*)d_in[9];
    const float* b_k  = (const float*)d_in[10];
    const float* w_o  = (const float*)d_in[11];
    const float* b_o  = (const float*)d_in[12];
    float* out   = (float*)d_out;
    float* scale = (float*)d_ws; // 101 floats of scratch

    scale_kernel<<<1, 128, 0, stream>>>(mb, w_up, b_up, scale);
    window_kernel<<<NWIN, 256, 0, stream>>>(attn, v, scale,
                                            w_q, b_q, w_k, b_k, w_o, b_o, out);
}